// Loss_65781719105930
// MI455X (gfx1250) — compile-verified
//
#include <hip/hip_runtime.h>
#include <math.h>

#define NC 20

typedef __attribute__((ext_vector_type(2))) float v2f_t;
typedef __attribute__((ext_vector_type(8))) float v8f_t;

#if __has_builtin(__builtin_amdgcn_wmma_f32_16x16x4_f32)
#define USE_WMMA_REDUCE 1
#else
#define USE_WMMA_REDUCE 0
#endif

// Exact f32 sum across all 32 lanes of the wave.
// A-layout (ISA 16x4 f32): lanes 0-15 -> A[m][0]=v0, A[m][1]=v1; lanes 16-31 -> A[m][2]=v0, A[m][3]=v1.
// With A = {p, 0} and B = all-ones: D[m][n] = p[m] + p[m+16] for every column n.
// Each lane holds 8 rows of one column of D; in-lane sum + xor16 fold = full 32-lane sum.
__device__ __forceinline__ float wave_sum32(float p) {
#if USE_WMMA_REDUCE
  v2f_t a; a.x = p; a.y = 0.0f;
  v2f_t ones; ones.x = 1.0f; ones.y = 1.0f;
  v8f_t c = {};
  c = __builtin_amdgcn_wmma_f32_16x16x4_f32(false, a, false, ones, (short)0, c, false, false);
  float s = ((c[0] + c[1]) + (c[2] + c[3])) + ((c[4] + c[5]) + (c[6] + c[7]));
  s += __shfl_xor(s, 16, 32);
  return s;
#else
  #pragma unroll
  for (int off = 16; off > 0; off >>= 1) p += __shfl_xor(p, off, 32);
  return p;
#endif
}

__global__ void __launch_bounds__(256)
yolo_v2_loss_kernel(const float* __restrict__ out,
                    const float* __restrict__ gt_boxes,
                    const float* __restrict__ anchors,
                    const int*   __restrict__ gt_classes,
                    const int*   __restrict__ num_box,
                    float*       __restrict__ loss,
                    int B, int H, int W, int A, int M)
{
  const int HW   = H * W;
  const int i    = blockIdx.x * blockDim.x + threadIdx.x;
  const int b    = i / HW;
  const int cell = i - b * HW;
  const int gy   = cell / W;
  const int gx   = cell - gy * W;

  __shared__ float s_box[64 * 4];
  __shared__ int   s_cls[64];
  __shared__ float s_anchor[16 * 2];
  __shared__ float s_part[8][4];

  // Stage this batch's GT data once per block (all 256 threads share one batch: 256 | H*W).
  for (int t = threadIdx.x; t < M * 4; t += blockDim.x)
    s_box[t] = gt_boxes[(size_t)b * M * 4 + t];
  for (int t = threadIdx.x; t < M; t += blockDim.x)
    s_cls[t] = gt_classes[(size_t)b * M + t];
  for (int t = threadIdx.x; t < A * 2; t += blockDim.x)
    s_anchor[t] = anchors[t];
  __syncthreads();

  // build_target, recomputed per cell: last matching box wins (sequential scatter
  // semantics); class target = argmax of accumulated one-hots = min class index.
  const int nb = num_box[b];
  float obj = 0.0f, tx = 0.0f, ty = 0.0f, tw = 0.0f, th = 0.0f;
  int cmin = NC;
  for (int m = 0; m < nb; ++m) {
    float x = s_box[m * 4 + 0] * (float)H;
    float y = s_box[m * 4 + 1] * (float)H;
    int cgx = (int)floorf(x);
    int cgy = (int)floorf(y);
    if (cgx == gx && cgy == gy) {
      obj = 1.0f;
      tx = x - (float)cgx; ty = y - (float)cgy;
      tw = s_box[m * 4 + 2] * (float)H;
      th = s_box[m * 4 + 3] * (float)H;
      cmin = min(cmin, s_cls[m]);
    }
  }

  const int CH = 5 + NC;
  const float* po = out + (size_t)b * (size_t)(A * CH) * HW + cell;

  float noobj = 0.0f, box_p = 0.0f, conf_p = 0.0f, cls_p = 0.0f;
  const float tbx = (float)gx + tx;
  const float tby = (float)gy + ty;
  float best_iou = -1.0f;
  int   best_a = 0;
  float bconf = 0.0f, bpx = 0.0f, bpy = 0.0f, bpw = 0.0f, bph = 0.0f;

  for (int a = 0; a < A; ++a) {
    const float* pa = po + (size_t)(a * CH) * HW;
    // conf logit: needed for every cell (noobj term). Coalesced across the wave.
    float pc = 1.0f / (1.0f + expf(-pa[(size_t)NC * HW]));
    noobj += pc * pc;
    if (obj > 0.0f) {  // rare path: only cells holding a GT box
      float px = (float)gx + 1.0f / (1.0f + expf(-pa[(size_t)(NC + 1) * HW]));
      float py = (float)gy + 1.0f / (1.0f + expf(-pa[(size_t)(NC + 2) * HW]));
      float pw = expf(pa[(size_t)(NC + 3) * HW]) * s_anchor[a * 2 + 0];
      float ph = expf(pa[(size_t)(NC + 4) * HW]) * s_anchor[a * 2 + 1];
      float ax1 = px - pw * 0.5f, ax2 = px + pw * 0.5f;
      float ay1 = py - ph * 0.5f, ay2 = py + ph * 0.5f;
      float bx1 = tbx - tw * 0.5f, bx2 = tbx + tw * 0.5f;
      float by1 = tby - th * 0.5f, by2 = tby + th * 0.5f;
      float iw = fmaxf(fminf(ax2, bx2) - fmaxf(ax1, bx1), 0.0f);
      float ih = fmaxf(fminf(ay2, by2) - fmaxf(ay1, by1), 0.0f);
      float inter = iw * ih;
      float uni = pw * ph + tw * th - inter;
      float iou = inter / fmaxf(uni, 1e-10f);
      if (iou > best_iou) {   // strict '>' matches argmax first-occurrence tie-break
        best_iou = iou; best_a = a; bconf = pc;
        bpx = px; bpy = py; bpw = pw; bph = ph;
      }
    }
  }

  if (obj > 0.0f) {
    conf_p = (bconf - 1.0f) * (bconf - 1.0f);
    noobj -= bconf * bconf;                 // mask removes best anchor from noobj sum
    float dx = bpx - tbx, dy = bpy - tby;
    float dw = sqrtf(bpw) - sqrtf(tw);
    float dh = sqrtf(bph) - sqrtf(th);
    box_p = dx * dx + dy * dy + dw * dw + dh * dh;
    // Cross-entropy on the best anchor's class logits only (2 passes, L1 hits on 2nd).
    const float* pl = po + (size_t)(best_a * CH) * HW;
    float mx = -3.0e38f;
    for (int j = 0; j < NC; ++j) mx = fmaxf(mx, pl[(size_t)j * HW]);
    float se = 0.0f, lc = 0.0f;
    for (int j = 0; j < NC; ++j) {
      float l = pl[(size_t)j * HW];
      se += expf(l - mx);
      if (j == cmin) lc = l;
    }
    cls_p = (mx + logf(se)) - lc;
  }

  const float invB = 1.0f / (float)B;
  float p0 = box_p  * (5.0f * invB);   // LAM_COORD / B
  float p1 = conf_p * (1.0f * invB);   // LAM_OBJ / B
  float p2 = noobj  * (0.5f * invB);   // LAM_NOOBJ / B
  float p3 = cls_p  * (1.0f * invB);   // LAM_CLS / B

  // All threads reach here (no early returns) => EXEC all ones for WMMA.
  float s0 = wave_sum32(p0);
  float s1 = wave_sum32(p1);
  float s2 = wave_sum32(p2);
  float s3 = wave_sum32(p3);

  const int lane = threadIdx.x & 31;
  const int wv   = threadIdx.x >> 5;
  if (lane == 0) {
    s_part[wv][0] = s0; s_part[wv][1] = s1;
    s_part[wv][2] = s2; s_part[wv][3] = s3;
  }
  __syncthreads();
  if (threadIdx.x < 4) {
    const int nw = blockDim.x >> 5;
    float t = 0.0f;
    for (int w = 0; w < nw; ++w) t += s_part[w][threadIdx.x];
    atomicAdd(&loss[threadIdx.x], t);
  }
}

extern "C" void kernel_launch(void* const* d_in, const int* in_sizes, int n_in,
                              void* d_out, int out_size, void* d_ws, size_t ws_size,
                              hipStream_t stream) {
  (void)n_in; (void)out_size; (void)d_ws; (void)ws_size;
  const float* out_t = (const float*)d_in[0];
  const float* gtb   = (const float*)d_in[1];
  const float* anc   = (const float*)d_in[2];
  const int*   gtc   = (const int*)d_in[3];
  const int*   nbx   = (const int*)d_in[4];

  const int B  = in_sizes[4];
  const int M  = in_sizes[3] / B;
  const int A  = in_sizes[2] / 2;
  const int HW = in_sizes[0] / (B * A * (5 + NC));
  const int W  = (int)(sqrtf((float)HW) + 0.5f);
  const int H  = HW / W;

  float* loss = (float*)d_out;
  hipMemsetAsync(loss, 0, 4 * sizeof(float), stream);  // graph-capturable memset node

  const int N = B * HW;            // 256*1024, divisible by 256
  const int threads = 256;         // 8 wave32s per block
  const int blocks  = N / threads;
  yolo_v2_loss_kernel<<<blocks, threads, 0, stream>>>(out_t, gtb, anc, gtc, nbx,
                                                      loss, B, H, W, A, M);
}